// CISGNN_58952721105384
// MI455X (gfx1250) — compile-verified
//
#include <hip/hip_runtime.h>

#define N_USERS 100000
#define N_ITEMS 200000
#define N_TOTAL (N_USERS + N_ITEMS)
#define D 64
#define NNZ_R 8000000
#define NNZ_S 2000000

typedef float v2f __attribute__((ext_vector_type(2)));
typedef float v8f __attribute__((ext_vector_type(8)));

// ---------------------------------------------------------------------------
// init: cur = out(acc) = concat(user_emb, item_emb). 4 floats / thread.
// ---------------------------------------------------------------------------
__global__ void k_init(const float* __restrict__ ue, const float* __restrict__ ie,
                       float* __restrict__ cur, float* __restrict__ out) {
  size_t base = ((size_t)blockIdx.x * blockDim.x + threadIdx.x) * 4;
  const size_t U = (size_t)N_USERS * D;           // 6.4M, multiple of 4
  if (base >= (size_t)N_TOTAL * D) return;
  float4 v = (base < U) ? *(const float4*)(ue + base)
                        : *(const float4*)(ie + (base - U));
  *(float4*)(cur + base) = v;
  *(float4*)(out + base) = v;
}

// ---------------------------------------------------------------------------
// zero a buffer (float4 granularity)
// ---------------------------------------------------------------------------
__global__ void k_zero(float* __restrict__ buf, size_t n4) {
  size_t i = (size_t)blockIdx.x * blockDim.x + threadIdx.x;
  if (i >= n4) return;
  *(float4*)(buf + i * 4) = make_float4(0.f, 0.f, 0.f, 0.f);
}

// ---------------------------------------------------------------------------
// COO SpMM scatter: TWO edges per wave32 (one per half-wave).
// Lane handles 4 dims: one b128 gather + 4 native global_atomic_add_f32.
// Edge stream prefetched ahead via global_prefetch_b8.
// ---------------------------------------------------------------------------
__global__ void k_spmm(const int* __restrict__ rows, const int* __restrict__ cols,
                       const float* __restrict__ vals, const float* __restrict__ x,
                       float* __restrict__ y, int nnz) {
  int wave = (int)(((size_t)blockIdx.x * blockDim.x + threadIdx.x) >> 5);
  int lane = threadIdx.x & 31;
  int e = wave * 2 + (lane >> 4);                 // edge for this half-wave
  if (e >= nnz) return;
  int dim0 = (lane & 15) * 4;                     // 4 dims per lane

  if (lane == 0) {                                // stream-ahead prefetch
    int pf = e + 8192;
    if (pf < nnz) {
      __builtin_prefetch(rows + pf, 0, 3);
      __builtin_prefetch(cols + pf, 0, 3);
      __builtin_prefetch(vals + pf, 0, 3);
    }
  }

  int r   = rows[e];
  int c   = cols[e];
  float v = vals[e];
  float4 xv = *(const float4*)(x + (size_t)c * D + dim0);
  float* yp = y + (size_t)r * D + dim0;
  unsafeAtomicAdd(yp + 0, v * xv.x);
  unsafeAtomicAdd(yp + 1, v * xv.y);
  unsafeAtomicAdd(yp + 2, v * xv.z);
  unsafeAtomicAdd(yp + 3, v * xv.w);
}

// ---------------------------------------------------------------------------
// acc += x, one 16x16 f32 tile per wave, via 4 chained
// V_WMMA_F32_16X16X4_F32 with shifted-identity B:
//   C <- A_j x B_j + C,  A_j = X[:,4j:4j+4],  B_j[k][n] = (n==4j+k)
// ---------------------------------------------------------------------------
__global__ void k_acc_wmma(const float* __restrict__ x, float* __restrict__ out,
                           int nTiles) {
  int wave = blockIdx.x * (blockDim.x >> 5) + (threadIdx.x >> 5);
  if (wave >= nTiles) return;                     // wave-uniform
  int lane = threadIdx.x & 31;
  int n  = lane & 15;                             // N index (and M for A-layout)
  int hi = lane >> 4;                             // lane-half selector
  int col0 = (wave & 3) * 16;                     // D/16 = 4 column tiles
  int row0 = (wave >> 2) * 16;

  // C/D layout: VGPR r holds M = r + 8*hi, N = n
  v8f c;
  const float* cbase = out + (size_t)(row0 + hi * 8) * D + col0 + n;
#pragma unroll
  for (int r = 0; r < 8; ++r) c[r] = cbase[(size_t)r * D];

#pragma unroll
  for (int j = 0; j < 4; ++j) {
    int k0 = j * 4 + hi * 2;                      // A: VGPR0/1 = K {k0, k0+1}
    v2f a = *(const v2f*)(x + (size_t)(row0 + n) * D + col0 + k0);
    v2f b;                                        // shifted identity
    b.x = (n == k0)     ? 1.0f : 0.0f;
    b.y = (n == k0 + 1) ? 1.0f : 0.0f;
    c = __builtin_amdgcn_wmma_f32_16x16x4_f32(false, a, false, b,
                                              (short)0, c, false, false);
  }

  float* obase = out + (size_t)(row0 + hi * 8) * D + col0 + n;
#pragma unroll
  for (int r = 0; r < 8; ++r) obase[(size_t)r * D] = c[r];
}

// ---------------------------------------------------------------------------
// Final combine: out = acc * 0.25 + (user rows only: ue/3 + (2/3)*s1)
// WMMA with B = 0.25*shifted-identity, C preloaded with the social term.
// User/item boundary (100000) is 16-row aligned -> branch is wave-uniform.
// ---------------------------------------------------------------------------
__global__ void k_final_wmma(float* __restrict__ out, const float* __restrict__ ue,
                             const float* __restrict__ s1, int nTiles) {
  int wave = blockIdx.x * (blockDim.x >> 5) + (threadIdx.x >> 5);
  if (wave >= nTiles) return;
  int lane = threadIdx.x & 31;
  int n  = lane & 15;
  int hi = lane >> 4;
  int col0 = (wave & 3) * 16;
  int row0 = (wave >> 2) * 16;

  const float inv3 = 1.0f / 3.0f;
  const float two3 = 2.0f / 3.0f;
  bool userTile = (row0 < N_USERS);               // tile-aligned boundary

  v8f c;
#pragma unroll
  for (int r = 0; r < 8; ++r) {
    float cv = 0.0f;
    if (userTile) {
      size_t off = (size_t)(row0 + hi * 8 + r) * D + col0 + n;
      cv = inv3 * ue[off] + two3 * s1[off];
    }
    c[r] = cv;
  }

#pragma unroll
  for (int j = 0; j < 4; ++j) {
    int k0 = j * 4 + hi * 2;
    v2f a = *(const v2f*)(out + (size_t)(row0 + n) * D + col0 + k0);
    v2f b;
    b.x = (n == k0)     ? 0.25f : 0.0f;
    b.y = (n == k0 + 1) ? 0.25f : 0.0f;
    c = __builtin_amdgcn_wmma_f32_16x16x4_f32(false, a, false, b,
                                              (short)0, c, false, false);
  }

  float* obase = out + (size_t)(row0 + hi * 8) * D + col0 + n;
#pragma unroll
  for (int r = 0; r < 8; ++r) obase[(size_t)r * D] = c[r];
}

// ---------------------------------------------------------------------------
extern "C" void kernel_launch(void* const* d_in, const int* in_sizes, int n_in,
                              void* d_out, int out_size, void* d_ws, size_t ws_size,
                              hipStream_t stream) {
  const float* ue = (const float*)d_in[0];
  const float* ie = (const float*)d_in[1];
  const float* Rv = (const float*)d_in[2];
  const float* Sv = (const float*)d_in[3];
  const int*   Rr = (const int*)d_in[4];
  const int*   Rc = (const int*)d_in[5];
  const int*   Sr = (const int*)d_in[6];
  const int*   Sc = (const int*)d_in[7];
  float* out = (float*)d_out;

  float* buf0 = (float*)d_ws;                       // N_TOTAL*D floats
  float* buf1 = buf0 + (size_t)N_TOTAL * D;         // N_TOTAL*D floats

  const size_t totalElems = (size_t)N_TOTAL * D;    // 19,200,000
  const int blk = 256;
  const int initBlocks = (int)((totalElems / 4 + blk - 1) / blk);   // 18750
  const int nTiles = (N_TOTAL / 16) * (D / 16);                     // 75000
  const int tileBlocks = (nTiles + (blk / 32) - 1) / (blk / 32);    // 9375

  // acc (in d_out) = cur = all_emb
  k_init<<<initBlocks, blk, 0, stream>>>(ue, ie, buf0, out);

  // 3 LightGCN layers on R (2 edges per wave -> 16 edges per 256-thr block)
  float* cur = buf0;
  float* nxt = buf1;
  for (int layer = 0; layer < 3; ++layer) {
    k_zero<<<initBlocks, blk, 0, stream>>>(nxt, totalElems / 4);
    k_spmm<<<NNZ_R / 16, blk, 0, stream>>>(Rr, Rc, Rv, cur, nxt, NNZ_R);
    k_acc_wmma<<<tileBlocks, blk, 0, stream>>>(nxt, out, nTiles);
    float* t = cur; cur = nxt; nxt = t;
  }

  // s1 = S @ user_emb  (reuse freed ping-pong buffer `nxt`)
  float* s1 = nxt;
  const size_t userElems = (size_t)N_USERS * D;     // 6,400,000
  k_zero<<<(int)(userElems / 4 / blk), blk, 0, stream>>>(s1, userElems / 4);
  k_spmm<<<NNZ_S / 16, blk, 0, stream>>>(Sr, Sc, Sv, ue, s1, NNZ_S);

  // out = acc/4 (+ social term on user rows)
  k_final_wmma<<<tileBlocks, blk, 0, stream>>>(out, ue, s1, nTiles);
}